// SignNetNodeEncoder_19619410608182
// MI455X (gfx1250) — compile-verified
//
#include <hip/hip_runtime.h>
#include <math.h>

// ---------------------------------------------------------------------------
// SignNet node encoder for MI455X (gfx1250).
// GIN stack collapsed via linearity: m_pm = +-A^3(p)(W0 W1) + A^2(b0)W1 + A(b1)
// => only scalar (K=8) edge aggregations + degree aggregations are needed.
// All dense layers run on V_WMMA_F32_16X16X4_F32 (fp32 matrix cores).
// ---------------------------------------------------------------------------

typedef float v2f __attribute__((ext_vector_type(2)));
typedef float v8f __attribute__((ext_vector_type(8)));

#define NN   50000     // nodes
#define KEIG 8         // eigenvectors
#define NE   800000    // edges
#define HID  64
#define OUTC 16
#define NK   (NN*KEIG) // 400000, multiple of 16
#define XC   112       // DIM_EMB - DIM_PE

static __device__ __forceinline__ v8f wmma_k4(v2f a, v2f b, v8f c) {
  // D = A(16x4 f32) * B(4x16 f32) + C(16x16 f32)
  return __builtin_amdgcn_wmma_f32_16x16x4_f32(false, a, false, b, (short)0, c,
                                               false, false);
}

// --- precompute small fused weight vectors (single block, 64 threads) -------
// sv[0:64)   wa = (W0·W1)·W2a        (coefficient of ±t)
// sv[64:128) q1 = b1·W2a             (coefficient of d1)
// sv[128:192)q2 = (b0·W1)·W2a        (coefficient of d2)
__global__ void precompute_vecs(const float* __restrict__ W0,
                                const float* __restrict__ b0,
                                const float* __restrict__ W1,
                                const float* __restrict__ b1,
                                const float* __restrict__ W2a,
                                float* __restrict__ sv) {
  __shared__ float w[HID], bw[HID];
  int j = threadIdx.x;                      // 64 threads
  float aw = 0.f, ab = 0.f;
  for (int h = 0; h < HID; ++h) {
    float w1 = W1[h * HID + j];
    aw += W0[h] * w1;                       // W0 is [1,64]
    ab += b0[h] * w1;
  }
  w[j] = aw; bw[j] = ab;
  __syncthreads();
  float wa = 0.f, q2 = 0.f, q1 = 0.f;
  for (int h = 0; h < HID; ++h) {
    float w2 = W2a[h * HID + j];
    wa += w[h] * w2;
    q2 += bw[h] * w2;
    q1 += b1[h] * w2;
  }
  sv[j] = wa; sv[64 + j] = q1; sv[128 + j] = q2;
}

// --- init: t = nan_to_num(eigvecs), zero scatter buffers --------------------
__global__ void init_t(const float* __restrict__ eig, float* __restrict__ t,
                       float* __restrict__ ts, float* __restrict__ ds) {
  int i = blockIdx.x * blockDim.x + threadIdx.x;
  if (i < NK) {
    float v = eig[i];
    v = (v == v) ? v : 0.f;                           // NaN -> 0
    v = fminf(fmaxf(v, -3.402823466e38f), 3.402823466e38f);  // clamp inf
    t[i] = v;
    ts[i] = 0.f;
  }
  if (i < NN) ds[i] = 0.f;
}

// --- per-edge scalar scatter: ts[dst,k] += t[src,k]; optional degree term ---
// mode 0: eig only; mode 1: also dacc[dst] += 1; mode 2: also dacc[dst] += dvec[src]
__global__ void edge_scatter(const int* __restrict__ src,
                             const int* __restrict__ dst,
                             const float* __restrict__ t,
                             float* __restrict__ ts,
                             const float* __restrict__ dvec,
                             float* __restrict__ dacc, int mode) {
  int e = blockIdx.x * blockDim.x + threadIdx.x;
  if (e >= NE) return;
  int s = src[e], d = dst[e];
  const float* tp = t + (long)s * KEIG;
  float* op = ts + (long)d * KEIG;
#pragma unroll
  for (int k = 0; k < KEIG; ++k) atomicAdd(op + k, tp[k]);
  if (mode == 1)      atomicAdd(dacc + d, 1.0f);
  else if (mode == 2) atomicAdd(dacc + d, dvec[s]);
}

// --- t += ts; ts = 0  (A = I + S) ------------------------------------------
__global__ void node_accum(float* __restrict__ t, float* __restrict__ ts) {
  int i = blockIdx.x * blockDim.x + threadIdx.x;
  if (i < NK) { t[i] += ts[i]; ts[i] = 0.f; }
}

__global__ void deg1_k(float* __restrict__ d1, float* __restrict__ ds) {
  int i = blockIdx.x * blockDim.x + threadIdx.x;
  if (i < NN) { d1[i] = 1.f + ds[i]; ds[i] = 0.f; }
}

__global__ void deg2_k(const float* __restrict__ d1, const float* __restrict__ ds,
                       float* __restrict__ d2) {
  int i = blockIdx.x * blockDim.x + threadIdx.x;
  if (i < NN) d2[i] = d1[i] + ds[i];
}

// --- fused sign-invariant ReLU + W2b GEMM via WMMA --------------------------
// rows = (node,k) pairs: s[row][j] = relu(+t·wa[j]+cw[j]) + relu(-t·wa[j]+cw[j])
// pe[node, k*16+n] = s @ W2b + 2*b2b.  One wave = one 16-row tile, 16 wmma.
__global__ void pe_w2b_wmma(const float* __restrict__ t3,
                            const float* __restrict__ d1,
                            const float* __restrict__ d2,
                            const float* __restrict__ sv,
                            const float* __restrict__ b2a,
                            const float* __restrict__ W2b,
                            const float* __restrict__ b2b,
                            float* __restrict__ pe) {
  __shared__ float lW[HID * OUTC];
  __shared__ float lwa[HID], lq1[HID], lq2[HID], lba[HID], lbb[OUTC];
  int tid = threadIdx.x;                    // 256 threads = 8 waves
  for (int i = tid; i < HID * OUTC; i += 256) lW[i] = W2b[i];
  if (tid < HID) {
    lwa[tid] = sv[tid]; lq1[tid] = sv[64 + tid]; lq2[tid] = sv[128 + tid];
    lba[tid] = b2a[tid];
  }
  if (tid < OUTC) lbb[tid] = b2b[tid];
  __syncthreads();

  int wave = tid >> 5, lane = tid & 31;
  int tile = blockIdx.x * 8 + wave;         // 25000 tiles exactly
  int m = lane & 15, half = lane >> 4;
  int row = tile * 16 + m;                  // (node,k) row
  int node = row >> 3;
  float tv = t3[row], d1v = d1[node], d2v = d2[node];

  v8f acc = {};
#pragma unroll
  for (int kk = 0; kk < 16; ++kk) {
    int j0 = kk * 4 + half * 2;             // A-frag K pair per ISA layout
    v2f a, b;
#pragma unroll
    for (int jj = 0; jj < 2; ++jj) {
      int j = j0 + jj;
      float cw = d2v * lq2[j] + d1v * lq1[j] + lba[j];
      float ev = tv * lwa[j];
      a[jj] = fmaxf(cw + ev, 0.f) + fmaxf(cw - ev, 0.f);  // relu(+)+relu(-)
      b[jj] = lW[j * OUTC + m];
    }
    acc = wmma_k4(a, b, acc);
  }
#pragma unroll
  for (int v = 0; v < 8; ++v) {             // D: M = v + 8*half, N = m
    int rM = tile * 16 + v + 8 * half;
    int nd = rM >> 3, kq = rM & 7;
    pe[(long)nd * 128 + kq * 16 + m] = acc[v] + 2.0f * lbb[m];
  }
}

// --- rho layer 1: z = relu(pe[N,128] @ Wr1[128,64] + br1) -------------------
// block = 128 threads = 4 waves; block -> one 16-row tile, wave -> 16-col tile
__global__ void rho1_wmma(const float* __restrict__ pe,
                          const float* __restrict__ Wr1,
                          const float* __restrict__ br1,
                          float* __restrict__ z) {
  __shared__ float lW[128 * 64];            // 32 KB (of 320 KB LDS)
  __shared__ float lb[64];
  int tid = threadIdx.x;
  for (int i = tid; i < 128 * 64; i += 128) lW[i] = Wr1[i];
  if (tid < 64) lb[tid] = br1[tid];
  __syncthreads();

  int wave = tid >> 5, lane = tid & 31;
  int rt = blockIdx.x, ct = wave;
  int m = lane & 15, half = lane >> 4;
  const float* ap = pe + (long)(rt * 16 + m) * 128;

  v8f acc = {};
#pragma unroll
  for (int kk = 0; kk < 32; ++kk) {
    int j0 = kk * 4 + half * 2;
    v2f a = { ap[j0], ap[j0 + 1] };
    v2f b = { lW[j0 * 64 + ct * 16 + m], lW[(j0 + 1) * 64 + ct * 16 + m] };
    acc = wmma_k4(a, b, acc);
  }
#pragma unroll
  for (int v = 0; v < 8; ++v) {
    int rM = rt * 16 + v + 8 * half;
    z[(long)rM * 64 + ct * 16 + m] = fmaxf(acc[v] + lb[ct * 16 + m], 0.f);
  }
}

// --- rho layer 2: out[:,112:128] = z[N,64] @ Wr2[64,16] + br2 ---------------
__global__ void rho2_wmma(const float* __restrict__ z,
                          const float* __restrict__ Wr2,
                          const float* __restrict__ br2,
                          float* __restrict__ out) {
  __shared__ float lW[HID * OUTC];
  __shared__ float lb[OUTC];
  int tid = threadIdx.x;                    // 256 = 8 waves = 8 row tiles
  for (int i = tid; i < HID * OUTC; i += 256) lW[i] = Wr2[i];
  if (tid < OUTC) lb[tid] = br2[tid];
  __syncthreads();

  int wave = tid >> 5, lane = tid & 31;
  int tile = blockIdx.x * 8 + wave;
  if (tile >= NN / 16) return;              // wave-uniform guard (3125 tiles)
  int m = lane & 15, half = lane >> 4;
  const float* ap = z + (long)(tile * 16 + m) * 64;

  v8f acc = {};
#pragma unroll
  for (int kk = 0; kk < 16; ++kk) {
    int j0 = kk * 4 + half * 2;
    v2f a = { ap[j0], ap[j0 + 1] };
    v2f b = { lW[j0 * OUTC + m], lW[(j0 + 1) * OUTC + m] };
    acc = wmma_k4(a, b, acc);
  }
#pragma unroll
  for (int v = 0; v < 8; ++v) {
    int rM = tile * 16 + v + 8 * half;
    out[(long)rM * 128 + 112 + m] = acc[v] + lb[m];
  }
}

// --- expand_x: out[:,0:112] = x[N,64] @ Wx[64,112] + bx ---------------------
// block = 256 = 8 waves; waves 0..6 -> 7 col tiles; one 16-row tile per block
__global__ void xlin_wmma(const float* __restrict__ x,
                          const float* __restrict__ Wx,
                          const float* __restrict__ bx,
                          float* __restrict__ out) {
  __shared__ float lW[HID * XC];            // 28 KB
  __shared__ float lb[XC];
  int tid = threadIdx.x;
  for (int i = tid; i < HID * XC; i += 256) lW[i] = Wx[i];
  if (tid < XC) lb[tid] = bx[tid];
  __syncthreads();

  int wave = tid >> 5, lane = tid & 31;
  if (wave >= 7) return;                    // wave-uniform
  int rt = blockIdx.x, ct = wave;
  int m = lane & 15, half = lane >> 4;
  const float* ap = x + (long)(rt * 16 + m) * 64;

  v8f acc = {};
#pragma unroll
  for (int kk = 0; kk < 16; ++kk) {
    int j0 = kk * 4 + half * 2;
    v2f a = { ap[j0], ap[j0 + 1] };
    v2f b = { lW[j0 * XC + ct * 16 + m], lW[(j0 + 1) * XC + ct * 16 + m] };
    acc = wmma_k4(a, b, acc);
  }
#pragma unroll
  for (int v = 0; v < 8; ++v) {
    int rM = rt * 16 + v + 8 * half;
    out[(long)rM * 128 + ct * 16 + m] = acc[v] + lb[ct * 16 + m];
  }
}

// ---------------------------------------------------------------------------
extern "C" void kernel_launch(void* const* d_in, const int* in_sizes, int n_in,
                              void* d_out, int out_size, void* d_ws, size_t ws_size,
                              hipStream_t stream) {
  const float* eig = (const float*)d_in[0];
  const float* x   = (const float*)d_in[1];
  const int*   ei  = (const int*)d_in[2];
  // d_in[3] = batch_index (unused by reference output)
  const float* W0  = (const float*)d_in[4];
  const float* b0  = (const float*)d_in[5];
  const float* W1  = (const float*)d_in[6];
  const float* b1  = (const float*)d_in[7];
  const float* W2a = (const float*)d_in[8];
  const float* b2a = (const float*)d_in[9];
  const float* W2b = (const float*)d_in[10];
  const float* b2b = (const float*)d_in[11];
  const float* Wr1 = (const float*)d_in[12];
  const float* br1 = (const float*)d_in[13];
  const float* Wr2 = (const float*)d_in[14];
  const float* br2 = (const float*)d_in[15];
  const float* Wx  = (const float*)d_in[16];
  const float* bx  = (const float*)d_in[17];
  float* out = (float*)d_out;

  const int* srcI = ei;
  const int* dstI = ei + NE;

  // workspace layout (floats): ~42 MB total
  float* w  = (float*)d_ws;
  float* t  = w;                 // NK
  float* ts = t  + NK;           // NK
  float* ds = ts + NK;           // NN
  float* d1 = ds + NN;           // NN
  float* d2 = d1 + NN;           // NN
  float* sv = d2 + NN;           // 192 (+pad)
  float* pe = sv + 256;          // NN*128
  float* z  = pe + (long)NN * 128; // NN*64

  const int TB = 256;
  const int gNK = (NK + TB - 1) / TB;   // covers NN too
  const int gE  = (NE + TB - 1) / TB;
  const int gN  = (NN + TB - 1) / TB;

  precompute_vecs<<<1, 64, 0, stream>>>(W0, b0, W1, b1, W2a, sv);
  init_t<<<gNK, TB, 0, stream>>>(eig, t, ts, ds);

  // A^1: t = A(p), d1 = A(1)
  edge_scatter<<<gE, TB, 0, stream>>>(srcI, dstI, t, ts, nullptr, ds, 1);
  node_accum<<<gNK, TB, 0, stream>>>(t, ts);
  deg1_k<<<gN, TB, 0, stream>>>(d1, ds);
  // A^2: t = A^2(p), d2 = A(d1)
  edge_scatter<<<gE, TB, 0, stream>>>(srcI, dstI, t, ts, d1, ds, 2);
  node_accum<<<gNK, TB, 0, stream>>>(t, ts);
  deg2_k<<<gN, TB, 0, stream>>>(d1, ds, d2);
  // A^3: t = A^3(p)
  edge_scatter<<<gE, TB, 0, stream>>>(srcI, dstI, t, ts, nullptr, nullptr, 0);
  node_accum<<<gNK, TB, 0, stream>>>(t, ts);

  // dense layers on matrix cores
  pe_w2b_wmma<<<NK / 16 / 8, TB, 0, stream>>>(t, d1, d2, sv, b2a, W2b, b2b, pe); // 3125
  rho1_wmma<<<NN / 16, 128, 0, stream>>>(pe, Wr1, br1, z);                        // 3125
  rho2_wmma<<<(NN / 16 + 7) / 8, TB, 0, stream>>>(z, Wr2, br2, out);              // 391
  xlin_wmma<<<NN / 16, TB, 0, stream>>>(x, Wx, bx, out);                          // 3125
}